// RoIAlign_54520314855548
// MI455X (gfx1250) — compile-verified
//
#include <hip/hip_runtime.h>
#include <cstdint>

// RoI-Align (single center sample per bin), CDNA5 / gfx1250.
//
// features [C=256, H=200, W=200] f32, rois [N,4] f32, out [N,256,7,7] f32.
// Roofline: ~41 MB feature reads (L2-resident after first touch) + ~50 MB
// output writes -> ~4us floor at 23.3 TB/s; FLOPs (~100M) are negligible.
// WMMA is deliberately not used: densifying the 2-4-nonzero interpolation
// matrix inflates read traffic >20x on a bandwidth-bound op. The CDNA5
// features that matter here: async global->LDS staging (ASYNCcnt path),
// LDS-broadcast descriptors, wave32 block sizing, paired b64 gathers,
// coalesced non-temporal stores.

#define ROI_OUT_H 7
#define ROI_OUT_W 7
#define ROI_NPTS  49      // 7*7
#define ROI_C     256
#define ROI_FH    200
#define ROI_FW    200

// 2 x f32 vector with 4-byte alignment: lets us issue one global_load_b64
// per bilinear row pair even though the base is only DWORD-aligned.
typedef float vf2_t __attribute__((ext_vector_type(2)));
typedef vf2_t vf2_a4 __attribute__((aligned(4)));

__global__ __launch_bounds__(256, 8)
void roi_align_kernel(const float* __restrict__ feat,
                      const float* __restrict__ rois,
                      float* __restrict__ out)
{
    __shared__ float4 s_roi;            // one roi row [x1,y1,x2,y2]
    __shared__ int2   s_base[ROI_NPTS]; // pair-load base offset for row0 / row1
    __shared__ float4 s_w[ROI_NPTS];    // weights (a0,b0,a1,b1) for the 2 pairs

    const int n   = blockIdx.x;
    const int tid = threadIdx.x;

    // ---- Phase 0: async-stage this block's roi (16B) into LDS -------------
    // gfx1250 GLOBAL_LOAD_ASYNC_TO_LDS_B128 (GV mode), tracked by ASYNCcnt.
    if (tid == 0) {
        uint32_t lds_addr = (uint32_t)(uintptr_t)(&s_roi);
        uint64_t gaddr    = (uint64_t)(uintptr_t)(rois + (size_t)n * 4);
        asm volatile("global_load_async_to_lds_b128 %0, %1, off"
                     :: "v"(lds_addr), "v"(gaddr)
                     : "memory");
        asm volatile("s_wait_asynccnt 0" ::: "memory");
    }
    __syncthreads();

    // ---- Phase 1: 49 threads build per-point bilinear descriptors ---------
    if (tid < ROI_NPTS) {
        const float4 r = s_roi;
        const float fx1 = r.x * (float)ROI_FW;
        const float fy1 = r.y * (float)ROI_FH;
        const float fx2 = r.z * (float)ROI_FW;
        const float fy2 = r.w * (float)ROI_FH;

        const float bh = (fy2 - fy1) * (1.0f / (float)ROI_OUT_H);
        const float bw = (fx2 - fx1) * (1.0f / (float)ROI_OUT_W);
        const bool valid = (bh > 0.0f) && (bw > 0.0f);

        const int oh = tid / ROI_OUT_W;
        const int ow = tid - oh * ROI_OUT_W;

        float y = fy1 + ((float)oh + 0.5f) * bh;
        float x = fx1 + ((float)ow + 0.5f) * bw;
        y = fminf(fmaxf(y, 0.0f), (float)(ROI_FH - 1));
        x = fminf(fmaxf(x, 0.0f), (float)(ROI_FW - 1));

        const float yf = floorf(y);
        const float xf = floorf(x);
        const int y0 = (int)yf;
        const int x0 = (int)xf;
        const int y1 = min(y0 + 1, ROI_FH - 1);   // == y0 when wy==0 (clamped)
        const float wy = y - yf;   // exactly 0 when y clamped at the edge
        const float wx = x - xf;   // exactly 0 when x clamped at the edge

        // Row-pair weights: row r contributes a_r*f[base_r] + b_r*f[base_r+1].
        float a0 = (1.0f - wy) * (1.0f - wx);   // w00
        float b0 = (1.0f - wy) * wx;            // w01 (0 when x clamped)
        float a1 = wy * (1.0f - wx);            // w10
        float b1 = wy * wx;                     // w11
        int base0 = y0 * ROI_FW + x0;
        int base1 = y1 * ROI_FW + x0;
        if (x0 == ROI_FW - 1) {
            // wx==0 here: shift the pair left so base+1 stays in-bounds and
            // move the (only nonzero) weight onto the second pair element.
            base0 -= 1; base1 -= 1;
            b0 = a0; a0 = 0.0f;
            b1 = a1; a1 = 0.0f;
        }
        if (!valid) { a0 = b0 = a1 = b1 = 0.0f; }

        s_base[tid] = make_int2(base0, base1);
        s_w[tid]    = make_float4(a0, b0, a1, b1);
    }
    __syncthreads();

    // ---- Phase 2: 256 threads x 49 iters cover all C*49 = 12544 outputs ----
    // e = tid + 256*i walks this roi's output linearly: stores perfectly
    // coalesced; each element needs just 2 b64 gathers (bilinear row pairs)
    // that hit L0/L2 with strong spatial reuse inside a roi.
    float* __restrict__ o = out + (size_t)n * (size_t)(ROI_C * ROI_NPTS);

#pragma unroll 4
    for (int i = 0; i < ROI_NPTS; ++i) {
        const unsigned e = (unsigned)tid + ((unsigned)i << 8);
        const unsigned c = e / (unsigned)ROI_NPTS;          // magic-mul, no div
        const unsigned p = e - c * (unsigned)ROI_NPTS;

        const int2   b = s_base[p];
        const float4 w = s_w[p];
        const float* __restrict__ fc = feat + (size_t)c * (size_t)(ROI_FH * ROI_FW);

        const vf2_a4 r0 = *(const vf2_a4*)(fc + b.x);   // f[y0, x0], f[y0, x0+1]
        const vf2_a4 r1 = *(const vf2_a4*)(fc + b.y);   // f[y1, x0], f[y1, x0+1]

        float v = w.w * r1.y;
        v = fmaf(w.z, r1.x, v);
        v = fmaf(w.y, r0.y, v);
        v = fmaf(w.x, r0.x, v);

        // Output is write-once: non-temporal store keeps it out of L2's way.
        __builtin_nontemporal_store(v, o + e);
    }
}

extern "C" void kernel_launch(void* const* d_in, const int* in_sizes, int n_in,
                              void* d_out, int out_size, void* d_ws, size_t ws_size,
                              hipStream_t stream) {
    const float* feat = (const float*)d_in[0];   // [256,200,200] f32
    const float* rois = (const float*)d_in[1];   // [N,4] f32
    float* out = (float*)d_out;                  // [N,256,7,7] f32

    const int n_rois = in_sizes[1] / 4;
    if (n_rois <= 0) return;

    roi_align_kernel<<<dim3((unsigned)n_rois), dim3(256), 0, stream>>>(feat, rois, out);
}